// MetaGAT_62775241998833
// MI455X (gfx1250) — compile-verified
//
#include <hip/hip_runtime.h>
#include <math.h>

#define EMBD 64

typedef __attribute__((ext_vector_type(2))) float v2f;
typedef __attribute__((ext_vector_type(8))) float v8f;

__device__ __forceinline__ float leaky01(float x) { return x > 0.0f ? x : 0.01f * x; }

// float atomic max via sign-split int/uint ordering trick (init must be -inf)
__device__ __forceinline__ void atomicMaxF32(float* addr, float val) {
  if (val >= 0.0f) atomicMax((int*)addr, __float_as_int(val));
  else             atomicMin((unsigned int*)addr, __float_as_uint(val));
}

// ---------------- init / scatter helpers ----------------
__global__ void fill_f32_kernel(float* __restrict__ p, float v, int n) {
  for (int t = blockIdx.x * blockDim.x + threadIdx.x; t < n; t += gridDim.x * blockDim.x)
    p[t] = v;
}
__global__ void fill_i32_kernel(int* __restrict__ p, int v, int n) {
  for (int t = blockIdx.x * blockDim.x + threadIdx.x; t < n; t += gridDim.x * blockDim.x)
    p[t] = v;
}
// slot[node] = batch row (duplicates: one winner; all dup rows read winner's slot)
__global__ void scatter_slot_kernel(const int* __restrict__ ids, int* __restrict__ slot, int n) {
  int t = blockIdx.x * blockDim.x + threadIdx.x;
  if (t < n) slot[ids[t]] = t;
}

// ---------------- per-node attention scores ----------------
// one wave per node: 32 lanes x float2 = one 64-float row; two dot products
__global__ void scores_kernel(const float* __restrict__ tab, int nNodes,
                              const float* __restrict__ wa1, const float* __restrict__ wa2,
                              float* __restrict__ out1, float* __restrict__ out2) {
  int wid  = (blockIdx.x * blockDim.x + threadIdx.x) >> 5;
  int lane = threadIdx.x & 31;
  if (wid >= nNodes) return;
  const float2* row = (const float2*)(tab + (size_t)wid * EMBD);
  float2 v  = row[lane];
  float2 w1 = ((const float2*)wa1)[lane];
  float2 w2 = ((const float2*)wa2)[lane];
  float d1 = v.x * w1.x + v.y * w1.y;
  float d2 = v.x * w2.x + v.y * w2.y;
  #pragma unroll
  for (int off = 16; off > 0; off >>= 1) {
    d1 += __shfl_down(d1, off, 32);
    d2 += __shfl_down(d2, off, 32);
  }
  if (lane == 0) { out1[wid] = d1; out2[wid] = d2; }
}

// ---------------- edge softmax passes (only edges whose dst is in batch) ----------------
__global__ void edge_max_kernel(const int* __restrict__ src, const int* __restrict__ dst,
                                const float* __restrict__ s_src, const float* __restrict__ s_dst,
                                const int* __restrict__ slot, float* __restrict__ m, int nE) {
  int e = blockIdx.x * blockDim.x + threadIdx.x;
  if (e >= nE) return;
  int d = dst[e];
  int s = slot[d];
  if (s < 0) return;
  float x = leaky01(s_src[src[e]] + s_dst[d]);
  atomicMaxF32(&m[s], x);
}

__global__ void edge_den_kernel(const int* __restrict__ src, const int* __restrict__ dst,
                                const float* __restrict__ s_src, const float* __restrict__ s_dst,
                                const int* __restrict__ slot, const float* __restrict__ m,
                                float* __restrict__ den, int nE) {
  int e = blockIdx.x * blockDim.x + threadIdx.x;
  if (e >= nE) return;
  int d = dst[e];
  int s = slot[d];
  if (s < 0) return;
  float x = leaky01(s_src[src[e]] + s_dst[d]);
  atomicAdd(&den[s], expf(x - m[s]));
}

// wave-per-edge: lanes cover the 64 feature dims (2 per lane), coalesced gather + f32 atomics
__global__ void edge_agg_kernel(const int* __restrict__ src, const int* __restrict__ dst,
                                const float* __restrict__ s_src, const float* __restrict__ s_dst,
                                const int* __restrict__ slot, const float* __restrict__ m,
                                const float* __restrict__ den, const float* __restrict__ feat,
                                float* __restrict__ hc, int nE) {
  int gwave  = (blockIdx.x * blockDim.x + threadIdx.x) >> 5;
  int lane   = threadIdx.x & 31;
  int nWaves = (gridDim.x * blockDim.x) >> 5;
  for (int e = gwave; e < nE; e += nWaves) {
    int d = dst[e];
    int s = slot[d];
    if (s < 0) continue;
    int sr = src[e];
    float x = leaky01(s_src[sr] + s_dst[d]);
    float alpha = expf(x - m[s]) / den[s];
    const float* f = feat + (size_t)sr * EMBD;
    float* h = hc + (size_t)s * EMBD;
    atomicAdd(&h[lane],      alpha * f[lane]);
    atomicAdd(&h[lane + 32], alpha * f[lane + 32]);
  }
}

// ---------------- fused dense stage: relu(x@Ws.T+bs) | relu(h@Wn.T+bn) -> relu(cat@Wfc.T)
// one wave per 16 batch rows; f32 WMMA 16x16x4; LDS padded to kill bank conflicts
__global__ __launch_bounds__(32)
void dense_kernel(const float* __restrict__ emb, const int* __restrict__ ids,
                  const int* __restrict__ slot, const float* __restrict__ hc,
                  const float* __restrict__ Ws, const float* __restrict__ bs,
                  const float* __restrict__ Wn, const float* __restrict__ bn,
                  const float* __restrict__ Wfc, float* __restrict__ out, int colOff) {
  __shared__ float ldsA[16][68];    // A tile (stride 68: 16 row-reads hit distinct banks)
  __shared__ float cat[16][132];    // concat [sf | nb] tile (stride 132)

  const int tile = blockIdx.x;
  const int l    = threadIdx.x;     // 0..31
  const int half = l >> 4;          // A/B operand: K = 4*ks + 2*half + j ; M/N = l&15
  const int lm   = l & 15;

  // ---------- stage 1: self path, A = emb[ids] ----------
  for (int idx = l; idx < 16 * EMBD; idx += 32) {
    int r = idx >> 6, c = idx & 63;
    int node = ids[tile * 16 + r];
    ldsA[r][c] = emb[(size_t)node * EMBD + c];
  }
  __syncthreads();

  v8f acc0 = {}, acc1 = {}, acc2 = {}, acc3 = {};
  #pragma unroll
  for (int ks = 0; ks < 16; ++ks) {
    int k = ks * 4 + half * 2;
    v2f a; a.x = ldsA[lm][k]; a.y = ldsA[lm][k + 1];
    #pragma unroll
    for (int n = 0; n < 4; ++n) {
      int col = n * 16 + lm;
      v2f b; b.x = Ws[col * 64 + k]; b.y = Ws[col * 64 + k + 1];
      v8f* acc = (n == 0) ? &acc0 : (n == 1) ? &acc1 : (n == 2) ? &acc2 : &acc3;
      *acc = __builtin_amdgcn_wmma_f32_16x16x4_f32(false, a, false, b, (short)0, *acc, false, false);
    }
  }
  // bias + relu; C layout: vgpr j -> row M = j + 8*half, col N = n*16 + lm
  #pragma unroll
  for (int n = 0; n < 4; ++n) {
    int col = n * 16 + lm;
    float bv = bs[col];
    v8f acc = (n == 0) ? acc0 : (n == 1) ? acc1 : (n == 2) ? acc2 : acc3;
    #pragma unroll
    for (int j = 0; j < 8; ++j) {
      float v = acc[j] + bv;
      cat[j + half * 8][col] = v > 0.0f ? v : 0.0f;
    }
  }
  __syncthreads();

  // ---------- stage 2: neighbor path, A = hc[slot[ids]] ----------
  for (int idx = l; idx < 16 * EMBD; idx += 32) {
    int r = idx >> 6, c = idx & 63;
    int s = slot[ids[tile * 16 + r]];
    ldsA[r][c] = hc[(size_t)s * EMBD + c];
  }
  __syncthreads();

  acc0 = (v8f){}; acc1 = (v8f){}; acc2 = (v8f){}; acc3 = (v8f){};
  #pragma unroll
  for (int ks = 0; ks < 16; ++ks) {
    int k = ks * 4 + half * 2;
    v2f a; a.x = ldsA[lm][k]; a.y = ldsA[lm][k + 1];
    #pragma unroll
    for (int n = 0; n < 4; ++n) {
      int col = n * 16 + lm;
      v2f b; b.x = Wn[col * 64 + k]; b.y = Wn[col * 64 + k + 1];
      v8f* acc = (n == 0) ? &acc0 : (n == 1) ? &acc1 : (n == 2) ? &acc2 : &acc3;
      *acc = __builtin_amdgcn_wmma_f32_16x16x4_f32(false, a, false, b, (short)0, *acc, false, false);
    }
  }
  #pragma unroll
  for (int n = 0; n < 4; ++n) {
    int col = n * 16 + lm;
    float bv = bn[col];
    v8f acc = (n == 0) ? acc0 : (n == 1) ? acc1 : (n == 2) ? acc2 : acc3;
    #pragma unroll
    for (int j = 0; j < 8; ++j) {
      float v = acc[j] + bv;
      cat[j + half * 8][64 + col] = v > 0.0f ? v : 0.0f;
    }
  }
  __syncthreads();

  // ---------- stage 3: out = relu(cat @ Wfc.T), K = 128 ----------
  acc0 = (v8f){}; acc1 = (v8f){}; acc2 = (v8f){}; acc3 = (v8f){};
  #pragma unroll
  for (int ks = 0; ks < 32; ++ks) {
    int k = ks * 4 + half * 2;
    v2f a; a.x = cat[lm][k]; a.y = cat[lm][k + 1];
    #pragma unroll
    for (int n = 0; n < 4; ++n) {
      int col = n * 16 + lm;
      v2f b; b.x = Wfc[col * 128 + k]; b.y = Wfc[col * 128 + k + 1];
      v8f* acc = (n == 0) ? &acc0 : (n == 1) ? &acc1 : (n == 2) ? &acc2 : &acc3;
      *acc = __builtin_amdgcn_wmma_f32_16x16x4_f32(false, a, false, b, (short)0, *acc, false, false);
    }
  }
  #pragma unroll
  for (int n = 0; n < 4; ++n) {
    int col = n * 16 + lm;
    #pragma unroll
    for (int j = 0; j < 8; ++j) {
      int row = tile * 16 + j + half * 8;
      v8f acc = (n == 0) ? acc0 : (n == 1) ? acc1 : (n == 2) ? acc2 : acc3;
      float v = acc[j];
      out[(size_t)row * 128 + colOff + col] = v > 0.0f ? v : 0.0f;
    }
  }
}

// ---------------- launch ----------------
extern "C" void kernel_launch(void* const* d_in, const int* in_sizes, int n_in,
                              void* d_out, int out_size, void* d_ws, size_t ws_size,
                              hipStream_t stream) {
  const float* user_emb = (const float*)d_in[0];
  const float* item_emb = (const float*)d_in[1];
  const float* Wa_u  = (const float*)d_in[2];
  const float* Wa_i  = (const float*)d_in[3];
  const float* Wfc_u = (const float*)d_in[4];
  const float* Wfc_i = (const float*)d_in[5];
  const float* Ws_u  = (const float*)d_in[6];
  const float* bs_u  = (const float*)d_in[7];
  const float* Ws_i  = (const float*)d_in[8];
  const float* bs_i  = (const float*)d_in[9];
  const float* Wn_u  = (const float*)d_in[10];
  const float* bn_u  = (const float*)d_in[11];
  const float* Wn_i  = (const float*)d_in[12];
  const float* bn_i  = (const float*)d_in[13];
  const int* u       = (const int*)d_in[14];
  const int* iv      = (const int*)d_in[15];
  const int* src_iu  = (const int*)d_in[16];
  const int* dst_iu  = (const int*)d_in[17];
  const int* src_ui  = (const int*)d_in[18];
  const int* dst_ui  = (const int*)d_in[19];

  const int NU = in_sizes[0] / EMBD;
  const int NI = in_sizes[1] / EMBD;
  const int B  = in_sizes[14];
  const int E  = in_sizes[16];

  // workspace carve-out
  char* w = (char*)d_ws;
  auto carve = [&](size_t bytes) { char* p = w; w += (bytes + 255) & ~(size_t)255; return p; };
  float* su_src = (float*)carve((size_t)NI * 4);   // item_emb @ Wa_u[:64]
  float* su_dst = (float*)carve((size_t)NU * 4);   // user_emb @ Wa_u[64:]
  float* si_src = (float*)carve((size_t)NU * 4);   // user_emb @ Wa_i[:64]
  float* si_dst = (float*)carve((size_t)NI * 4);   // item_emb @ Wa_i[64:]
  int*   slot_u = (int*)carve((size_t)NU * 4);
  int*   slot_i = (int*)carve((size_t)NI * 4);
  float* m_u    = (float*)carve((size_t)B * 4);
  float* den_u  = (float*)carve((size_t)B * 4);
  float* m_i    = (float*)carve((size_t)B * 4);
  float* den_i  = (float*)carve((size_t)B * 4);
  float* hc_u   = (float*)carve((size_t)B * EMBD * 4);
  float* hc_i   = (float*)carve((size_t)B * EMBD * 4);

  const int T = 256;
  auto blks = [](int n, int t) { return (n + t - 1) / t; };

  // init (every call: determinism)
  fill_i32_kernel<<<blks(NU, T), T, 0, stream>>>(slot_u, -1, NU);
  fill_i32_kernel<<<blks(NI, T), T, 0, stream>>>(slot_i, -1, NI);
  fill_f32_kernel<<<blks(B, T), T, 0, stream>>>(m_u, -INFINITY, B);
  fill_f32_kernel<<<blks(B, T), T, 0, stream>>>(m_i, -INFINITY, B);
  fill_f32_kernel<<<blks(B, T), T, 0, stream>>>(den_u, 0.0f, B);
  fill_f32_kernel<<<blks(B, T), T, 0, stream>>>(den_i, 0.0f, B);
  fill_f32_kernel<<<blks(B * EMBD, T), T, 0, stream>>>(hc_u, 0.0f, B * EMBD);
  fill_f32_kernel<<<blks(B * EMBD, T), T, 0, stream>>>(hc_i, 0.0f, B * EMBD);
  scatter_slot_kernel<<<blks(B, T), T, 0, stream>>>(u, slot_u, B);
  scatter_slot_kernel<<<blks(B, T), T, 0, stream>>>(iv, slot_i, B);

  // per-node scores (one wave per node -> 8 nodes per 256-thread block)
  scores_kernel<<<blks(NI * 32, T), T, 0, stream>>>(item_emb, NI, Wa_u, Wa_i + EMBD, su_src, si_dst);
  scores_kernel<<<blks(NU * 32, T), T, 0, stream>>>(user_emb, NU, Wa_u + EMBD, Wa_i, su_dst, si_src);

  // user side: items -> users (dst = users)
  edge_max_kernel<<<blks(E, T), T, 0, stream>>>(src_iu, dst_iu, su_src, su_dst, slot_u, m_u, E);
  edge_den_kernel<<<blks(E, T), T, 0, stream>>>(src_iu, dst_iu, su_src, su_dst, slot_u, m_u, den_u, E);
  edge_agg_kernel<<<4096, T, 0, stream>>>(src_iu, dst_iu, su_src, su_dst, slot_u, m_u, den_u,
                                          item_emb, hc_u, E);
  // item side: users -> items (dst = items)
  edge_max_kernel<<<blks(E, T), T, 0, stream>>>(src_ui, dst_ui, si_src, si_dst, slot_i, m_i, E);
  edge_den_kernel<<<blks(E, T), T, 0, stream>>>(src_ui, dst_ui, si_src, si_dst, slot_i, m_i, den_i, E);
  edge_agg_kernel<<<4096, T, 0, stream>>>(src_ui, dst_ui, si_src, si_dst, slot_i, m_i, den_i,
                                          user_emb, hc_i, E);

  // dense WMMA stage: [B,64] each side into out[B,128]
  float* out = (float*)d_out;
  dense_kernel<<<B / 16, 32, 0, stream>>>(user_emb, u, slot_u, hc_u, Ws_u, bs_u, Wn_u, bn_u,
                                          Wfc_u, out, 0);
  dense_kernel<<<B / 16, 32, 0, stream>>>(item_emb, iv, slot_i, hc_i, Ws_i, bs_i, Wn_i, bn_i,
                                          Wfc_i, out, EMBD);
}